// GraphAttentionConvolution_72670846648729
// MI455X (gfx1250) — compile-verified
//
#include <hip/hip_runtime.h>

// ---------------------------------------------------------------------------
// GAT layer, fused for MI455X (gfx1250).
//   Roofline: adj stream (256 MB) ~11us at 23.3 TB/s is the floor; the 8192^2
//   attention matrix is never materialized (flash-attention style online
//   softmax). Wh (2 MB) is L2-resident; WhT tiles are TDM-staged into LDS and
//   shared by a 4-wave group to cut L2 re-read traffic 4x.
//   Matrix math: V_WMMA_F32_16X16X4_F32 (full fp32 precision).
//   Data movement: TDM tensor_load_to_lds, 4-buffer pipeline (3 tiles ahead),
//   LDS pad feature (2 dwords per 32 -> stride 34) keeps ds_load_b64 aligned
//   and bank-conflict-free. 1024 waves: each 16-row block is split across two
//   waves (even/odd tiles) and merged at the end through LDS.
// ---------------------------------------------------------------------------

#define NN    8192
#define FIN   512
#define FOUT  64
#define ALPHA 0.2f
#define MASKV -9000000000000000.0f

#define TC    32          // columns (neighbors) per tile
#define TSTR  34          // padded LDS row stride in dwords (32 data + 2 pad)
#define RB    16          // rows per wave (WMMA M)
#define PD    4           // pipeline depth (buffers)
#define NJ    128         // tiles per wave (NN/TC/2, two-way column split)

#define ADJB  (RB * TSTR * 4)     // adj buffer bytes  (2176)
#define WHTB  (FOUT * TSTR * 4)   // wht buffer bytes  (8704)

typedef float    v2f __attribute__((ext_vector_type(2)));
typedef float    v8f __attribute__((ext_vector_type(8)));
typedef unsigned v4u __attribute__((ext_vector_type(4)));
typedef int      v4i __attribute__((ext_vector_type(4)));
typedef int      v8i __attribute__((ext_vector_type(8)));

// ---------------------------------------------------------------------------
// TDM: 2D tile load (4-byte elements) global -> LDS, padded 2 dwords per 32.
// Descriptor per CDNA5 ISA 8.3/8.4. Tracked on TENSORcnt, EXEC ignored.
// ---------------------------------------------------------------------------
__device__ __forceinline__ void tdm_load_2d(unsigned lds_off, const void* gptr,
                                            unsigned tile_d0, unsigned tile_d1,
                                            unsigned tensor_d0, unsigned tensor_d1,
                                            unsigned long long stride0)
{
    unsigned long long ga = (unsigned long long)gptr;
    v4u g0;
    g0.x = 1u;                                             // count=1 (valid)
    g0.y = lds_off;                                        // LDS byte address
    g0.z = (unsigned)ga;                                   // global addr [31:0]
    g0.w = (unsigned)((ga >> 32) & 0x01FFFFFFull)          // global addr [56:32]
         | 0x80000000u;                                    // type=2 ("image")
    v8i g1;
    g1[0] = (int)((2u << 16)                               // data_size = 4B
                | (1u << 20)                               // pad_enable
                | (4u << 22)                               // pad_interval: 32 dw
                | (1u << 25));                             // pad_amount: 2 dw
    g1[1] = (int)((tensor_d0 & 0xFFFFu) << 16);            // tensor_dim0 lo
    g1[2] = (int)(((tensor_d0 >> 16) & 0xFFFFu)
                | ((tensor_d1 & 0xFFFFu) << 16));          // dim0 hi | dim1 lo
    g1[3] = (int)(((tensor_d1 >> 16) & 0xFFFFu)
                | (tile_d0 << 16));                        // dim1 hi | tile_dim0
    g1[4] = (int)(tile_d1 & 0xFFFFu);                      // tile_dim1, tile_dim2=0
    g1[5] = (int)(stride0 & 0xFFFFFFFFull);                // tensor_dim0_stride lo
    g1[6] = (int)((stride0 >> 32) & 0xFFFFull);            // stride hi | dim1_stride=0
    g1[7] = 0;
    v4i z4 = {0, 0, 0, 0};
#if defined(__clang_major__) && __clang_major__ >= 23
    v8i z8 = {0, 0, 0, 0, 0, 0, 0, 0};
    __builtin_amdgcn_tensor_load_to_lds(g0, g1, z4, z4, z8, 0);
#else
    __builtin_amdgcn_tensor_load_to_lds(g0, g1, z4, z4, 0);
#endif
}

// ---------------------------------------------------------------------------
// Kernel 1a: WhT[n][i] = sum_k x[i][k] * W[k][n] (transposed so the fused
// kernel's WMMA B operand is contiguous). x loads are wave-uniform; W loads
// coalesced.
// ---------------------------------------------------------------------------
__global__ __launch_bounds__(256) void gat_wh_kernel(
    const float* __restrict__ x, const float* __restrict__ W,
    float* __restrict__ whT)
{
    const int gid = blockIdx.x * 256 + threadIdx.x;
    const int i = gid >> 6;
    const int n = gid & 63;
    const float* xr = x + (size_t)i * FIN;
    float acc = 0.0f;
    for (int k = 0; k < FIN; k += 4) {
        const float4 xv = *(const float4*)(xr + k);
        acc += xv.x * W[(k + 0) * FOUT + n];
        acc += xv.y * W[(k + 1) * FOUT + n];
        acc += xv.z * W[(k + 2) * FOUT + n];
        acc += xv.w * W[(k + 3) * FOUT + n];
    }
    whT[(size_t)n * NN + i] = acc;
}

// ---------------------------------------------------------------------------
// Kernel 1b: s1 = Wh @ a1, s2 = Wh @ a2. Coalesced over nodes.
// ---------------------------------------------------------------------------
__global__ __launch_bounds__(256) void gat_s_kernel(
    const float* __restrict__ whT, const float* __restrict__ A,
    float* __restrict__ s1, float* __restrict__ s2)
{
    const int i = blockIdx.x * 256 + threadIdx.x;
    float a1s = 0.0f, a2s = 0.0f;
    for (int n = 0; n < FOUT; ++n) {
        const float w = whT[(size_t)n * NN + i];
        a1s += w * A[n];
        a2s += w * A[FOUT + n];
    }
    s1[i] = a1s;
    s2[i] = a2s;
}

// ---------------------------------------------------------------------------
// Kernel 2: fused masked softmax + aggregation.
//   Block = 256 threads = 8 waves = 2 schedule groups of 4 waves.
//   Group 0 handles even tiles, group 1 odd tiles, for the same 4 row blocks;
//   partial (m, l, acc) merged through LDS at the end.
//   Per group, wave (grp*4) is the leader: it TDM-loads the shared WhT tile;
//   every wave TDM-loads its private adj tile. One __syncthreads per tile
//   publishes the shared tile and licenses buffer reuse (4-deep pipeline).
// ---------------------------------------------------------------------------
__global__ __launch_bounds__(256) void gat_attn_kernel(
    const int* __restrict__ adj, const float* __restrict__ whT,
    const float* __restrict__ s1, const float* __restrict__ s2,
    const float* __restrict__ bias, float* __restrict__ out)
{
    __shared__ int   adjS[8][PD][RB * TSTR];     // 68 KB (reused for merge)
    __shared__ float whtS[2][PD][FOUT * TSTR];   // 68 KB

    const int wave = threadIdx.x >> 5;
    const int lane = threadIdx.x & 31;
    const int r = lane & 15;            // matrix row within 16-row block
    const int h = lane >> 4;            // half-wave selector (K pair)
    const int grp  = wave >> 2;         // schedule group: tile parity
    const int rbl  = wave & 3;          // local row block
    const bool leader = (rbl == 0);
    const int rowbase = (blockIdx.x * 4 + rbl) * RB;

    const float s1r = s1[rowbase + r];
    const float b0v = bias[r], b1v = bias[16 + r], b2v = bias[32 + r], b3v = bias[48 + r];

    // LDS byte offsets (low 32 bits of a generic shared pointer == LDS offset)
    const unsigned adjOffBase = (unsigned)(unsigned long long)(const void*)&adjS[wave][0][0];
    const unsigned whtOffBase = (unsigned)(unsigned long long)(const void*)&whtS[grp][0][0];

    v8f acc0 = {}, acc1 = {}, acc2 = {}, acc3 = {};
    float mrun = MASKV;   // running row max (MASKV not -inf: all-masked rows stay NaN-free)
    float lrun = 0.0f;    // running row sum

    const size_t adjRowBase = (size_t)rowbase * NN;

    // Prologue: prefetch tiles j = 0,1,2 (global tile index tj = 2*j + grp).
    for (int p = 0; p < PD - 1; ++p) {
        const int tj = 2 * p + grp;
        tdm_load_2d(adjOffBase + p * ADJB, adj + adjRowBase + (size_t)tj * TC,
                    TC, RB, NN, NN, NN);
        if (leader)
            tdm_load_2d(whtOffBase + p * WHTB, whT + (size_t)tj * TC,
                        TC, FOUT, NN, FOUT, NN);
    }

    for (int j = 0; j < NJ; ++j) {
        const int buf = j & (PD - 1);
        const int tj = 2 * j + grp;

        // 1) Ensure this tile's TDM data landed. In-order TENSORcnt:
        //    leader keeps 2 [adj,wht] pairs in flight (<=4), others 2 adj (<=2).
        if (j < NJ - 2) {
            if (leader) __builtin_amdgcn_s_wait_tensorcnt(4);
            else        __builtin_amdgcn_s_wait_tensorcnt(2);
        } else {
            __builtin_amdgcn_s_wait_tensorcnt(0);
        }
        // 2) Publish shared WhT tile; license overwrite of buffer (j+3)%4
        //    (last read during iteration j-1 by every wave in the block).
        __syncthreads();

        // 3) Prefetch tile j+3.
        if (j + PD - 1 < NJ) {
            const int tn = 2 * (j + PD - 1) + grp;
            const int nb = (j + PD - 1) & (PD - 1);
            asm volatile("s_wait_dscnt 0x0" ::: "memory");
            tdm_load_2d(adjOffBase + nb * ADJB, adj + adjRowBase + (size_t)tn * TC,
                        TC, RB, NN, NN, NN);
            if (leader)
                tdm_load_2d(whtOffBase + nb * WHTB, whT + (size_t)tn * TC,
                            TC, FOUT, NN, FOUT, NN);
        }

        // 4) Compute tile tj from buffer `buf`.
        const int*   __restrict__ aT  = &adjS[wave][buf][0];
        const float* __restrict__ wT  = &whtS[grp][buf][0];
        const float* __restrict__ s2t = s2 + tj * TC;

        // masked leaky-relu logits for this lane's 16 columns (WMMA A pattern)
        float lg[16];
        float mloc = MASKV;
#pragma unroll
        for (int s = 0; s < 8; ++s) {
            const int c = 4 * s + 2 * h;
            const int2   av = *(const int2*)(aT + r * TSTR + c);
            const float2 sv = *(const float2*)(s2t + c);
            float z0 = s1r + sv.x; z0 = (z0 > 0.0f) ? z0 : (ALPHA * z0);
            float z1 = s1r + sv.y; z1 = (z1 > 0.0f) ? z1 : (ALPHA * z1);
            z0 = (av.x > 0) ? z0 : MASKV;
            z1 = (av.y > 0) ? z1 : MASKV;
            lg[2 * s] = z0; lg[2 * s + 1] = z1;
            mloc = fmaxf(mloc, fmaxf(z0, z1));
        }

        // online softmax update (lane pair r / r+16 shares a row)
        mloc = fmaxf(mloc, __shfl_xor(mloc, 16, 32));
        const float mnew = fmaxf(mrun, mloc);
        const float corr = __expf(mrun - mnew);
        mrun = mnew;

        float pv[16];
        float ps = 0.0f;
#pragma unroll
        for (int q = 0; q < 16; ++q) { pv[q] = __expf(lg[q] - mnew); ps += pv[q]; }
        ps += __shfl_xor(ps, 16, 32);
        lrun = lrun * corr + ps;

        // rescale accumulators: C-layout row m = v + 8h; corr[m] lives in lane m
#pragma unroll
        for (int v = 0; v < 8; ++v) {
            const float cf = __shfl(corr, v + 8 * h, 32);
            acc0[v] *= cf; acc1[v] *= cf; acc2[v] *= cf; acc3[v] *= cf;
        }

        // acc += P(16xTC) @ WhTile(TCx64) via fp32 WMMA
#pragma unroll
        for (int s = 0; s < 8; ++s) {
            const int c = 4 * s + 2 * h;
            v2f a; a.x = pv[2 * s]; a.y = pv[2 * s + 1];
            const v2f bb0 = *(const v2f*)(wT + (0 * 16 + r) * TSTR + c);
            const v2f bb1 = *(const v2f*)(wT + (1 * 16 + r) * TSTR + c);
            const v2f bb2 = *(const v2f*)(wT + (2 * 16 + r) * TSTR + c);
            const v2f bb3 = *(const v2f*)(wT + (3 * 16 + r) * TSTR + c);
            acc0 = __builtin_amdgcn_wmma_f32_16x16x4_f32(false, a, false, bb0, (short)0, acc0, false, false);
            acc1 = __builtin_amdgcn_wmma_f32_16x16x4_f32(false, a, false, bb1, (short)0, acc1, false, false);
            acc2 = __builtin_amdgcn_wmma_f32_16x16x4_f32(false, a, false, bb2, (short)0, acc2, false, false);
            acc3 = __builtin_amdgcn_wmma_f32_16x16x4_f32(false, a, false, bb3, (short)0, acc3, false, false);
        }
    }

    // ---- merge the two column-halves of each row block through LDS ----
    __syncthreads();                       // all TDM/ds traffic drained above
    float* M = ((float*)&adjS[0][0][0]) + rbl * (16 * 64 + 32);
    if (grp == 1) {                        // odd-tile wave dumps its partials
        if (lane < 16) { M[16 * 64 + r] = mrun; M[16 * 64 + 16 + r] = lrun; }
#pragma unroll
        for (int v = 0; v < 8; ++v) {
            const int m = v + 8 * h;
            M[m * 64 +      r] = acc0[v];
            M[m * 64 + 16 + r] = acc1[v];
            M[m * 64 + 32 + r] = acc2[v];
            M[m * 64 + 48 + r] = acc3[v];
        }
    }
    __syncthreads();
    if (grp == 0) {                        // even-tile wave combines + writes
        const float mo = M[16 * 64 + r];
        const float lo = M[16 * 64 + 16 + r];
        const float mm = fmaxf(mrun, mo);
        const float fe = __expf(mrun - mm);
        const float fo = __expf(mo - mm);
        const float linv = 1.0f / (lrun * fe + lo * fo);
#pragma unroll
        for (int v = 0; v < 8; ++v) {
            const float sfe = __shfl(fe, v + 8 * h, 32);
            const float sfo = __shfl(fo, v + 8 * h, 32);
            const float sli = __shfl(linv, v + 8 * h, 32);
            const int m = v + 8 * h;
            float* o = out + (size_t)(rowbase + m) * FOUT;
            o[r]      = (acc0[v] * sfe + M[m * 64 +      r] * sfo) * sli + b0v;
            o[16 + r] = (acc1[v] * sfe + M[m * 64 + 16 + r] * sfo) * sli + b1v;
            o[32 + r] = (acc2[v] * sfe + M[m * 64 + 32 + r] * sfo) * sli + b2v;
            o[48 + r] = (acc3[v] * sfe + M[m * 64 + 48 + r] * sfo) * sli + b3v;
        }
    }
}

// ---------------------------------------------------------------------------
extern "C" void kernel_launch(void* const* d_in, const int* in_sizes, int n_in,
                              void* d_out, int out_size, void* d_ws, size_t ws_size,
                              hipStream_t stream)
{
    (void)in_sizes; (void)n_in; (void)out_size; (void)ws_size;
    const float* x    = (const float*)d_in[0];
    const int*   adj  = (const int*)d_in[1];
    const float* W    = (const float*)d_in[2];
    const float* bias = (const float*)d_in[3];
    const float* A    = (const float*)d_in[4];
    float* out = (float*)d_out;

    float* whT = (float*)d_ws;               // 64 x 8192 f32 (2 MB)
    float* s1  = whT + (size_t)FOUT * NN;    // 8192 f32
    float* s2  = s1 + NN;                    // 8192 f32

    gat_wh_kernel<<<(NN * FOUT) / 256, 256, 0, stream>>>(x, W, whT);
    gat_s_kernel<<<NN / 256, 256, 0, stream>>>(whT, A, s1, s2);
    gat_attn_kernel<<<(NN / RB) / 4, 256, 0, stream>>>(adj, whT, s1, s2, bias, out);
}